// xLSTM_AE_29197187678700
// MI455X (gfx1250) — compile-verified
//
#include <hip/hip_runtime.h>
#include <hip/hip_bf16.h>
#include <cstdint>

// ---------------- model dims ----------------
#define B_    128
#define S_    256
#define D_    128
#define H_    4
#define HD_   32
#define LAT_  32
#define DFF_  192
#define NB_   4
#define NTOK  (B_ * S_)   // 32768 tokens

typedef float v2f __attribute__((ext_vector_type(2)));
typedef float v8f __attribute__((ext_vector_type(8)));
typedef unsigned int u32x4 __attribute__((ext_vector_type(4)));
typedef int i32x4 __attribute__((ext_vector_type(4)));
typedef int i32x8 __attribute__((ext_vector_type(8)));

// ---------------- fp32 WMMA 16x16x4 wrapper ----------------
// A: 16x4 (M=lane%16; VGPR0 -> K = 0/2 by half-wave, VGPR1 -> K = 1/3)
// B: 4x16 (N=lane%16; VGPR0 -> K = 0/2 by half-wave, VGPR1 -> K = 1/3)
// C/D: 16x16 (VGPR i -> row i (lanes 0-15) / row i+8 (lanes 16-31), col = lane%16)
__device__ __forceinline__ v8f wmma4(v2f a, v2f b, v8f c) {
  return __builtin_amdgcn_wmma_f32_16x16x4_f32(false, a, false, b, (short)0, c,
                                               false, false);
}

__device__ __forceinline__ float sigmoidf_(float x) {
  return 1.f / (1.f + __expf(-x));
}
__device__ __forceinline__ float logsigf_(float x) {
  return fminf(x, 0.f) - log1pf(__expf(-fabsf(x)));
}

// ---------------- elementwise / reduction kernels ----------------

__global__ void inproj_kernel(const float* __restrict__ x,
                              const float* __restrict__ w,
                              const float* __restrict__ b,
                              float* __restrict__ h) {
  size_t idx = (size_t)blockIdx.x * blockDim.x + threadIdx.x;
  if (idx >= (size_t)NTOK * D_) return;
  int d = (int)(idx & (D_ - 1));
  size_t t = idx >> 7;
  h[idx] = x[t] * w[d] + b[d];
}

// LayerNorm over D=128 per token. block = 128 threads, one token per block.
__global__ void ln_kernel(const float* __restrict__ in,
                          const float* __restrict__ w,
                          float* __restrict__ out) {
  size_t t = blockIdx.x;
  int d = threadIdx.x;
  float x = in[t * D_ + d];
  float s = x, s2 = x * x;
  for (int o = 16; o > 0; o >>= 1) {  // wave32 reduce
    s += __shfl_xor(s, o, 32);
    s2 += __shfl_xor(s2, o, 32);
  }
  __shared__ float ssum[4], ssum2[4];
  if ((d & 31) == 0) { ssum[d >> 5] = s; ssum2[d >> 5] = s2; }
  __syncthreads();
  s  = ssum[0] + ssum[1] + ssum[2] + ssum[3];
  s2 = ssum2[0] + ssum2[1] + ssum2[2] + ssum2[3];
  float mu  = s * (1.f / D_);
  float var = s2 * (1.f / D_) - mu * mu;
  out[t * D_ + d] = (x - mu) * rsqrtf(var + 1e-5f) * w[d];
}

// depthwise causal conv (K=4) + swish
__global__ void conv_swish_kernel(const float* __restrict__ xn,
                                  const float* __restrict__ cw,
                                  const float* __restrict__ cb,
                                  float* __restrict__ xc) {
  size_t idx = (size_t)blockIdx.x * blockDim.x + threadIdx.x;
  if (idx >= (size_t)NTOK * D_) return;
  int d = (int)(idx & (D_ - 1));
  size_t t = idx >> 7;
  int s = (int)(t % S_);
  size_t b = t / S_;
  float acc = cb[d];
#pragma unroll
  for (int j = 0; j < 4; ++j) {
    int sp = s - 3 + j;
    if (sp >= 0) acc += xn[(b * S_ + sp) * D_ + d] * cw[j * D_ + d];
  }
  xc[idx] = acc * sigmoidf_(acc);
}

// headwise gate projections via WMMA:
//   wx[s][b][h][g*32+k] = sum_d src[b,s,h*32+d] * w_g[h][d][k]
// grid = (NTOK/16, H, 4 gates); block = 32 (one wave)
__global__ void headwise_kernel(const float* __restrict__ xc,
                                const float* __restrict__ xn,
                                const float* __restrict__ wi,
                                const float* __restrict__ wf,
                                const float* __restrict__ wz,
                                const float* __restrict__ wo,
                                float* __restrict__ wx) {
  int lane = threadIdx.x;
  int tile = blockIdx.x, h = blockIdx.y, g = blockIdx.z;
  const float* src = (g < 2) ? xc : xn;  // i,f <- conv path; z,o <- ln path
  const float* w = (g == 0) ? wi : (g == 1) ? wf : (g == 2) ? wz : wo;
  w += (size_t)h * HD_ * HD_;            // w[h][d][k]
  int lm = lane & 15;
  int khalf = (lane >> 4) << 1;
  size_t tok = (size_t)tile * 16 + lm;
  v8f acc0 = {}, acc1 = {};
  for (int k0 = 0; k0 < HD_; k0 += 4) {
    v2f a, b0, b1;
    a.x = src[tok * D_ + h * HD_ + k0 + khalf + 0];
    a.y = src[tok * D_ + h * HD_ + k0 + khalf + 1];
    b0.x = w[(k0 + khalf + 0) * HD_ + lm];
    b0.y = w[(k0 + khalf + 1) * HD_ + lm];
    b1.x = w[(k0 + khalf + 0) * HD_ + 16 + lm];
    b1.y = w[(k0 + khalf + 1) * HD_ + 16 + lm];
    acc0 = wmma4(a, b0, acc0);
    acc1 = wmma4(a, b1, acc1);
  }
#pragma unroll
  for (int i = 0; i < 8; ++i) {
    size_t trow = (size_t)tile * 16 + i + ((lane >> 4) << 3);
    int s = (int)(trow % S_);
    size_t b = trow / S_;
    size_t base = (((size_t)s * B_ + b) * H_ + h) * (4 * HD_) + g * HD_;
    wx[base + lm]      = acc0[i];
    wx[base + 16 + lm] = acc1[i];
  }
}

// ----------------- the sequential sLSTM scan -----------------
// grid = B/16 workgroups; block = 128 threads = 4 waves, wave w handles head w.
// R (64 KB, all heads) is staged once into LDS -- via the Tensor Data Mover
// when the toolchain exposes it -- then reused for 256 steps; y feeds back
// through LDS; c,n,m state lives in VGPRs.
__global__ void scan_kernel(const float* __restrict__ wx,
                            const float* __restrict__ R,
                            const float* __restrict__ bias,
                            float* __restrict__ ys) {
  __shared__ float Rsh[H_][HD_][4 * HD_];  // 64 KB (of 320 KB/WGP)
  __shared__ float ysh[H_][16][HD_];       // 8 KB
  int tid = threadIdx.x;
  int h = tid >> 5;          // wave id == head
  int lane = tid & 31;
  int lm = lane & 15;
  int hi = lane >> 4;        // half-wave: 0/1
  int b0 = blockIdx.x * 16;

  // zero initial y
  {
    float* yh = &ysh[h][0][0];
    for (int i = lane; i < 16 * HD_; i += 32) yh[i] = 0.f;
  }

  // ---- stage R into LDS (TDM if available, else cooperative load) ----
#if __has_builtin(__builtin_amdgcn_tensor_load_to_lds)
  if (tid < 32) {  // one wave issues the DMA
    uint64_t gaddr = (uint64_t)(uintptr_t)R;
    uint32_t ldsoff = (uint32_t)(uintptr_t)&Rsh[0][0][0];
    // D# group0: count=1 | lds_addr | global_addr | type=2
    u32x4 g0 = {1u, ldsoff, (uint32_t)gaddr,
                (uint32_t)((gaddr >> 32) & 0x01FFFFFFu) | 0x80000000u};
    // D# group1: data_size=4B; tensor 2048x8 f32, tile 2048x8, stride 2048
    i32x8 g1 = {0x00020000,        // data_size=2 (4 bytes)
                0x08000000,        // tensor_dim0 low16 = 2048 at [31:16]
                0x00080000,        // tensor_dim1 low16 = 8 at [31:16]
                0x08000000,        // tile_dim0 = 2048 at [31:16]
                0x00000008,        // tile_dim1 = 8
                2048, 0, 0};       // tensor_dim0_stride = 2048
    i32x4 gz = {0, 0, 0, 0};
#if __clang_major__ >= 23
    i32x8 gz8 = {0, 0, 0, 0, 0, 0, 0, 0};
    __builtin_amdgcn_tensor_load_to_lds(g0, g1, gz, gz, gz8, 0);
#else
    __builtin_amdgcn_tensor_load_to_lds(g0, g1, gz, gz, 0);
#endif
#if __has_builtin(__builtin_amdgcn_s_wait_tensorcnt)
    __builtin_amdgcn_s_wait_tensorcnt(0);
#endif
  }
#else
  for (int i = tid; i < H_ * HD_ * 4 * HD_; i += 128)
    (&Rsh[0][0][0])[i] = R[i];
#endif
  __syncthreads();

  // state regs: index si = kh*8+v -> batch = b0 + v + hi*8, k = kh*16 + lm
  float c_[16], n_[16], m_[16];
#pragma unroll
  for (int i = 0; i < 16; ++i) { c_[i] = 0.f; n_[i] = 0.f; m_[i] = 0.f; }

  // bias per output-tile nt: col = nt*16+lm -> g = col/32, k = col%32
  float biasr[8];
#pragma unroll
  for (int nt = 0; nt < 8; ++nt) {
    int g = nt >> 1, k = (nt & 1) * 16 + lm;
    biasr[nt] = bias[(g * H_ + h) * HD_ + k];
  }

  const float* RL = &Rsh[h][0][0];

  for (int t = 0; t < S_; ++t) {
    v8f acc[8];
#pragma unroll
    for (int nt = 0; nt < 8; ++nt) acc[nt] = (v8f){};

    // ry = y @ R : K = 32 in 8 steps of 4, N = 128 in 8 tiles of 16 (LDS B-ops)
    for (int kk = 0; kk < 8; ++kk) {
      v2f a;
      a.x = ysh[h][lm][kk * 4 + hi * 2 + 0];
      a.y = ysh[h][lm][kk * 4 + hi * 2 + 1];
#pragma unroll
      for (int nt = 0; nt < 8; ++nt) {
        v2f b;
        b.x = RL[(kk * 4 + hi * 2 + 0) * (4 * HD_) + nt * 16 + lm];
        b.y = RL[(kk * 4 + hi * 2 + 1) * (4 * HD_) + nt * 16 + lm];
        acc[nt] = wmma4(a, b, acc[nt]);
      }
    }

    // prefetch next step's gate pre-activations (global_prefetch_b8)
    if (t + 1 < S_) {
      size_t pf =
          (((size_t)(t + 1) * B_ + (b0 + hi * 8)) * H_ + h) * (4 * HD_) + lm;
      __builtin_prefetch(&wx[pf], 0, 1);
    }

    // gate math for the 16 (batch,k) elements owned by this lane
#pragma unroll
    for (int kh = 0; kh < 2; ++kh) {
#pragma unroll
      for (int v = 0; v < 8; ++v) {
        int si = kh * 8 + v;
        int batch = b0 + v + hi * 8;
        size_t wb =
            (((size_t)t * B_ + batch) * H_ + h) * (4 * HD_) + kh * 16 + lm;
        float iraw = acc[0 + kh][v] + wx[wb +  0] + biasr[0 + kh];
        float fraw = acc[2 + kh][v] + wx[wb + 32] + biasr[2 + kh];
        float zraw = acc[4 + kh][v] + wx[wb + 64] + biasr[4 + kh];
        float oraw = acc[6 + kh][v] + wx[wb + 96] + biasr[6 + kh];
        float lfm  = m_[si] + logsigf_(fraw);
        float mnew = (t == 0) ? iraw : fmaxf(iraw, lfm);
        float og = sigmoidf_(oraw);
        float ig = __expf(iraw - mnew);
        float fg = __expf(lfm - mnew);
        float cnew = fg * c_[si] + ig * tanhf(zraw);
        float nnew = fg * n_[si] + ig;
        float ynew = og * cnew / nnew;
        c_[si] = cnew; n_[si] = nnew; m_[si] = mnew;
        ysh[h][v + hi * 8][kh * 16 + lm] = ynew;
        ys[((size_t)batch * S_ + t) * D_ + h * HD_ + kh * 16 + lm] = ynew;
      }
    }
    __syncthreads();
  }
}

// per-head LN (over HD=32) + residual add into h
__global__ void mhln_res_kernel(const float* __restrict__ ysrc,
                                const float* __restrict__ gn,
                                float* __restrict__ h) {
  size_t t = blockIdx.x;
  int d = threadIdx.x;  // d = head*32 + k
  float x = ysrc[t * D_ + d];
  float s = x, s2 = x * x;
  for (int o = 16; o > 0; o >>= 1) {
    s += __shfl_xor(s, o, 32);
    s2 += __shfl_xor(s2, o, 32);
  }
  float mu  = s * (1.f / HD_);
  float var = s2 * (1.f / HD_) - mu * mu;
  h[t * D_ + d] += (x - mu) * rsqrtf(var + 1e-5f) * gn[d];
}

// generic fp32 WMMA GEMM: C[M x N] = A[M x K] @ W[K x N] (+bias) (+residual)
// grid = (M/16, N/(16*NT)); block = 32 (one wave). A tile reused across NT
// column tiles -> 1/NT the A traffic; weights are L2-resident.
template <int NT>
__global__ void gemm_wmma_kernel(const float* __restrict__ A,
                                 const float* __restrict__ W,
                                 const float* __restrict__ bias,
                                 const float* __restrict__ res,
                                 float* __restrict__ C,
                                 int K, int lda, int ldw, int ldc) {
  int lane = threadIdx.x;
  int lm = lane & 15;
  int khalf = (lane >> 4) << 1;
  size_t mbase = (size_t)blockIdx.x * 16;
  int nbase = blockIdx.y * (16 * NT);
  size_t mrow = mbase + lm;
  v8f acc[NT];
#pragma unroll
  for (int nt = 0; nt < NT; ++nt) acc[nt] = (v8f){};
  for (int k0 = 0; k0 < K; k0 += 4) {
    v2f a;
    a.x = A[mrow * lda + k0 + khalf + 0];
    a.y = A[mrow * lda + k0 + khalf + 1];
#pragma unroll
    for (int nt = 0; nt < NT; ++nt) {
      v2f b;
      int ncol = nbase + nt * 16 + lm;
      b.x = W[(size_t)(k0 + khalf + 0) * ldw + ncol];
      b.y = W[(size_t)(k0 + khalf + 1) * ldw + ncol];
      acc[nt] = wmma4(a, b, acc[nt]);
    }
  }
#pragma unroll
  for (int nt = 0; nt < NT; ++nt) {
    int ncol = nbase + nt * 16 + lm;
    float bv = bias ? bias[ncol] : 0.f;
#pragma unroll
    for (int i = 0; i < 8; ++i) {
      size_t row = mbase + i + ((lane >> 4) << 3);
      size_t idx = row * ldc + ncol;
      float v = acc[nt][i] + bv;
      if (res) v += res[idx];
      C[idx] = v;
    }
  }
}

// gelu(tanh approx)(u[:, :192]) * u[:, 192:384]
__global__ void geglu_kernel(const float* __restrict__ u,
                             float* __restrict__ gv) {
  size_t idx = (size_t)blockIdx.x * blockDim.x + threadIdx.x;
  if (idx >= (size_t)NTOK * DFF_) return;
  int j = (int)(idx % DFF_);
  size_t t = idx / DFF_;
  float g = u[t * (2 * DFF_) + j];
  float v = u[t * (2 * DFF_) + DFF_ + j];
  float g3 = g + 0.044715f * g * g * g;
  float ge = 0.5f * g * (1.f + tanhf(0.7978845608028654f * g3));
  gv[idx] = ge * v;
}

// out[t] = dot(h[t,:], out_w) + out_b   (one wave per token)
__global__ void outproj_kernel(const float* __restrict__ h,
                               const float* __restrict__ w,
                               const float* __restrict__ b,
                               float* __restrict__ out) {
  size_t t = blockIdx.x;
  int lane = threadIdx.x;
  float s = 0.f;
  for (int d = lane; d < D_; d += 32) s += h[t * D_ + d] * w[d];
  for (int o = 16; o > 0; o >>= 1) s += __shfl_xor(s, o, 32);
  if (lane == 0) out[t] = s + b[0];
}

// ---------------- host-side orchestration ----------------
struct BlockP {
  const float *ln1, *conv_w, *conv_b, *wi, *wf, *wz, *wo, *R, *b, *gn, *ln2,
      *ffn_up, *ffn_down;
};

static BlockP get_block(void* const* din, int base, int i) {
  BlockP p;
  p.ln1      = (const float*)din[base + 0]  + (size_t)i * D_;
  p.conv_w   = (const float*)din[base + 1]  + (size_t)i * 4 * D_;
  p.conv_b   = (const float*)din[base + 2]  + (size_t)i * D_;
  p.wi       = (const float*)din[base + 3]  + (size_t)i * H_ * HD_ * HD_;
  p.wf       = (const float*)din[base + 4]  + (size_t)i * H_ * HD_ * HD_;
  p.wz       = (const float*)din[base + 5]  + (size_t)i * H_ * HD_ * HD_;
  p.wo       = (const float*)din[base + 6]  + (size_t)i * H_ * HD_ * HD_;
  p.R        = (const float*)din[base + 7]  + (size_t)i * H_ * HD_ * 4 * HD_;
  p.b        = (const float*)din[base + 8]  + (size_t)i * 4 * H_ * HD_;
  p.gn       = (const float*)din[base + 9]  + (size_t)i * H_ * HD_;
  p.ln2      = (const float*)din[base + 10] + (size_t)i * D_;
  p.ffn_up   = (const float*)din[base + 11] + (size_t)i * D_ * 2 * DFF_;
  p.ffn_down = (const float*)din[base + 12] + (size_t)i * DFF_ * D_;
  return p;
}

static void run_block(hipStream_t stream, float* h, float* t0, float* t1,
                      float* ys, float* wx, float* u, float* gv,
                      const BlockP& p) {
  const int EW = 256;
  ln_kernel<<<NTOK, D_, 0, stream>>>(h, p.ln1, t0);
  conv_swish_kernel<<<((size_t)NTOK * D_ + EW - 1) / EW, EW, 0, stream>>>(
      t0, p.conv_w, p.conv_b, t1);
  headwise_kernel<<<dim3(NTOK / 16, H_, 4), 32, 0, stream>>>(
      t1, t0, p.wi, p.wf, p.wz, p.wo, wx);
  scan_kernel<<<B_ / 16, 128, 0, stream>>>(wx, p.R, p.b, ys);
  mhln_res_kernel<<<NTOK, D_, 0, stream>>>(ys, p.gn, h);
  ln_kernel<<<NTOK, D_, 0, stream>>>(h, p.ln2, t0);
  // u = xn2 @ ffn_up (128 -> 384), 16x64 tiles per wave
  gemm_wmma_kernel<4><<<dim3(NTOK / 16, (2 * DFF_) / 64), 32, 0, stream>>>(
      t0, p.ffn_up, nullptr, nullptr, u, D_, D_, 2 * DFF_, 2 * DFF_);
  geglu_kernel<<<((size_t)NTOK * DFF_ + EW - 1) / EW, EW, 0, stream>>>(u, gv);
  // h = h + gv @ ffn_down (192 -> 128)
  gemm_wmma_kernel<4><<<dim3(NTOK / 16, D_ / 64), 32, 0, stream>>>(
      gv, p.ffn_down, nullptr, h, h, DFF_, DFF_, D_, D_);
}

extern "C" void kernel_launch(void* const* d_in, const int* in_sizes, int n_in,
                              void* d_out, int out_size, void* d_ws,
                              size_t ws_size, hipStream_t stream) {
  (void)in_sizes; (void)n_in; (void)out_size; (void)ws_size;
  const float* x        = (const float*)d_in[0];
  const float* in_w     = (const float*)d_in[1];
  const float* in_b     = (const float*)d_in[2];
  // enc dict: indices 3..15
  const float* enc_post = (const float*)d_in[16];
  const float* bneck_w  = (const float*)d_in[17];
  const float* bneck_b  = (const float*)d_in[18];
  const float* unb_w    = (const float*)d_in[19];
  const float* unb_b    = (const float*)d_in[20];
  // dec dict: indices 21..33
  const float* dec_post = (const float*)d_in[34];
  const float* out_w    = (const float*)d_in[35];
  const float* out_b    = (const float*)d_in[36];
  float* out = (float*)d_out;

  float* W = (float*)d_ws;
  const size_t TD = (size_t)NTOK * D_;
  float* h   = W;
  float* t0  = h + TD;
  float* t1  = t0 + TD;
  float* ys  = t1 + TD;
  float* wx  = ys + TD;                      // 16.8M floats (scan phase)
  float* u   = wx;                           // aliases wx (FFN phase)
  float* gv  = u + (size_t)NTOK * 2 * DFF_;
  float* lat = wx + ((size_t)NTOK * 2 * DFF_ + (size_t)NTOK * DFF_);
  const int EW = 256;

  inproj_kernel<<<(TD + EW - 1) / EW, EW, 0, stream>>>(x, in_w, in_b, h);

  for (int i = 0; i < NB_; ++i) {
    BlockP p = get_block(d_in, 3, i);
    run_block(stream, h, t0, t1, ys, wx, u, gv, p);
  }
  ln_kernel<<<NTOK, D_, 0, stream>>>(h, enc_post, h);

  // bottleneck: lat = h @ bneck_w + bneck_b ; h = lat @ unb_w + unb_b
  gemm_wmma_kernel<2><<<dim3(NTOK / 16, 1), 32, 0, stream>>>(
      h, bneck_w, bneck_b, nullptr, lat, D_, D_, LAT_, LAT_);
  gemm_wmma_kernel<4><<<dim3(NTOK / 16, D_ / 64), 32, 0, stream>>>(
      lat, unb_w, unb_b, nullptr, h, LAT_, LAT_, D_, D_);

  for (int i = 0; i < NB_; ++i) {
    BlockP p = get_block(d_in, 21, i);
    run_block(stream, h, t0, t1, ys, wx, u, gv, p);
  }
  ln_kernel<<<NTOK, D_, 0, stream>>>(h, dec_post, h);

  outproj_kernel<<<NTOK, 32, 0, stream>>>(h, out_w, out_b, out);
}